// DDIM_85040352460871
// MI455X (gfx1250) — compile-verified
//
#include <hip/hip_runtime.h>
#include <math.h>

// ---------------------------------------------------------------------------
// DDIM block for MI455X (gfx1250, wave32).
//  - All 1x1 convs / projections -> fp32 WMMA GEMM (v_wmma_f32_16x16x4_f32).
//  - Selective scan: 1024-thread blocks, 3 (d,n) states/thread, shfl_xor(16)
//    reduction over the N=16 state dim (wave32 subgroups).
//  - Everything stays resident in the 192MB L2 (total scratch ~75MB fp32).
// ---------------------------------------------------------------------------

constexpr int kB = 2, kC = 96, kD = 192, kN = 16, kH = 48, kW = 48;
constexpr int kL = kH * kW;      // 2304 = 36*64
constexpr int kXR = 38;          // R + 2N rows of x_proj
constexpr int kXRp = 48;         // padded to 3 WMMA M-tiles
constexpr int kRp = 8;           // dt rank 6 padded to 8 (K multiple of 4)

typedef __attribute__((ext_vector_type(2))) float v2f;
typedef __attribute__((ext_vector_type(8))) float v8f;

#define CDIV(a, b) (((a) + (b) - 1) / (b))

__device__ __forceinline__ float sigm(float x) { return 1.0f / (1.0f + expf(-x)); }
__device__ __forceinline__ float silu(float x) { return x * sigm(x); }
__device__ __forceinline__ float softplusf(float x) { return (x > 20.0f) ? x : log1pf(expf(x)); }

// ---------------------------------------------------------------------------
// WMMA GEMM: Y[z] (MxN) = W[z%KW] (MxK) * X[z] (KxN) (+bias) (+=accum)
// One wave computes a 16x64 tile. Requires M%16==0, N%64==0, K%4==0.
// A frag (16x4 f32): lanes 0-15 row=l16 K=kk..kk+1, lanes 16-31 K=kk+2..kk+3.
// B frag (4x16 f32): vgpr0 = row kk(+2 for hi lanes), vgpr1 = next row.
// C/D: vgpr r -> row r (lanes<16) / r+8 (lanes>=16), col = lane%16.
// ---------------------------------------------------------------------------
__global__ __launch_bounds__(32) void gemm_wmma_f32(
    const float* __restrict__ Wm, const float* __restrict__ X,
    const float* __restrict__ bias, float* __restrict__ Y,
    int M, int K, int N, int KW,
    long wstride, long xstride, long ystride, int accum) {
  (void)M;
  const int lane = threadIdx.x;
  const int half = lane >> 4;
  const int l16 = lane & 15;
  const int n0 = blockIdx.x * 64;
  const int m0 = blockIdx.y * 16;
  const int z = blockIdx.z;

  const float* Wp = Wm + (size_t)(z % KW) * wstride;
  const float* Xp = X + (size_t)z * xstride;
  float* Yp = Y + (size_t)z * ystride;

  v8f a0 = {}, a1 = {}, a2 = {}, a3 = {};
  for (int kk = 0; kk < K; kk += 4) {
    const float* wr = Wp + (size_t)(m0 + l16) * K + (kk + half * 2);
    v2f av;
    av.x = wr[0];
    av.y = wr[1];
    const float* xr = Xp + (size_t)(kk + half * 2) * N + n0 + l16;
    v2f b;
    b.x = xr[0];
    b.y = xr[N];
    a0 = __builtin_amdgcn_wmma_f32_16x16x4_f32(false, av, false, b, (short)0, a0, false, false);
    b.x = xr[16];
    b.y = xr[N + 16];
    a1 = __builtin_amdgcn_wmma_f32_16x16x4_f32(false, av, false, b, (short)0, a1, false, false);
    b.x = xr[32];
    b.y = xr[N + 32];
    a2 = __builtin_amdgcn_wmma_f32_16x16x4_f32(false, av, false, b, (short)0, a2, false, false);
    b.x = xr[48];
    b.y = xr[N + 48];
    a3 = __builtin_amdgcn_wmma_f32_16x16x4_f32(false, av, false, b, (short)0, a3, false, false);
  }
  for (int r = 0; r < 8; ++r) {
    const int row = m0 + r + half * 8;
    const float bv = bias ? bias[row] : 0.0f;
    float* yr = Yp + (size_t)row * N + n0 + l16;
    yr[0] = (accum ? yr[0] : 0.0f) + a0[r] + bv;
    yr[16] = (accum ? yr[16] : 0.0f) + a1[r] + bv;
    yr[32] = (accum ? yr[32] : 0.0f) + a2[r] + bv;
    yr[48] = (accum ? yr[48] : 0.0f) + a3[r] + bv;
  }
}

// LayerNorm over channel dim: X (Z, CH, L) -> Y, per (z,l).
__global__ void layernorm_rows(const float* __restrict__ X, float* __restrict__ Y,
                               const float* __restrict__ w, const float* __restrict__ b,
                               int CH, int L, int Z) {
  int idx = blockIdx.x * blockDim.x + threadIdx.x;
  if (idx >= Z * L) return;
  int z = idx / L, l = idx - z * L;
  const float* xp = X + (size_t)z * CH * L + l;
  float s = 0.f, s2 = 0.f;
  for (int c = 0; c < CH; ++c) {
    float v = xp[(size_t)c * L];
    s += v;
    s2 += v * v;
  }
  float m = s / CH;
  float var = s2 / CH - m * m;
  float inv = rsqrtf(var + 1e-5f);
  float* yp = Y + (size_t)z * CH * L + l;
  for (int c = 0; c < CH; ++c) {
    float v = xp[(size_t)c * L];
    yp[(size_t)c * L] = (v - m) * inv * w[c] + b[c];
  }
}

// Per-row (over L) mean and optional max. gridDim.x = B*rows_per_b.
__global__ void rowstat(const float* __restrict__ X, float* __restrict__ mean,
                        float* __restrict__ mx, int L, int rows_per_b, long bstride, long rstride) {
  int r = blockIdx.x;
  int b = r / rows_per_b;
  int rr = r - b * rows_per_b;
  const float* xp = X + (size_t)b * bstride + (size_t)rr * rstride;
  __shared__ float ssum[256];
  __shared__ float smax[256];
  float s = 0.f, mv = -3.4e38f;
  for (int i = threadIdx.x; i < L; i += 256) {
    float v = xp[i];
    s += v;
    mv = fmaxf(mv, v);
  }
  ssum[threadIdx.x] = s;
  smax[threadIdx.x] = mv;
  __syncthreads();
  for (int o = 128; o > 0; o >>= 1) {
    if (threadIdx.x < o) {
      ssum[threadIdx.x] += ssum[threadIdx.x + o];
      smax[threadIdx.x] = fmaxf(smax[threadIdx.x], smax[threadIdx.x + o]);
    }
    __syncthreads();
  }
  if (threadIdx.x == 0) {
    mean[r] = ssum[0] / L;
    if (mx) mx[r] = smax[0];
  }
}

// Per-pixel mean/max over CH channels (stride kL between channels).
__global__ void colstat(const float* __restrict__ X, float* __restrict__ outm,
                        float* __restrict__ outx, int CH, long bstride, long obstride, int total) {
  int idx = blockIdx.x * blockDim.x + threadIdx.x;
  if (idx >= total) return;
  int b = idx / kL, l = idx - b * kL;
  const float* xp = X + (size_t)b * bstride + l;
  float s = 0.f, m = -3.4e38f;
  for (int c = 0; c < CH; ++c) {
    float v = xp[(size_t)c * kL];
    s += v;
    m = fmaxf(m, v);
  }
  outm[(size_t)b * obstride + l] = s / CH;
  outx[(size_t)b * obstride + l] = m;
}

// CCA gate MLP: blockIdx.x = b*2 + which (0: fc(se) gates si, 1: fc(si) gates se)
__global__ void cca_gate(const float* __restrict__ gi, const float* __restrict__ ge,
                         const float* __restrict__ w1, const float* __restrict__ w2,
                         float* __restrict__ att) {
  int b = blockIdx.x >> 1, which = blockIdx.x & 1;
  int c = threadIdx.x;
  __shared__ float g[kC];
  __shared__ float hid[24];
  int odd = c & 1;
  float vi = gi[b * kC + c], ve = ge[b * kC + c];
  // se: odd->img, even->ev ; si: odd->ev, even->img
  g[c] = (which == 0) ? (odd ? vi : ve) : (odd ? ve : vi);
  __syncthreads();
  if (c < 24) {
    float s = 0.f;
    for (int i = 0; i < kC; ++i) s += w1[c * kC + i] * g[i];
    hid[c] = fmaxf(s, 0.f);
  }
  __syncthreads();
  float s = 0.f;
  for (int j = 0; j < 24; ++j) s += w2[c * 24 + j] * hid[j];
  att[(which * kB + b) * kC + c] = sigm(s);
}

// CSA per-pixel 4->16->2 MLP.
__global__ void csa_att(const float* __restrict__ stat, const float* __restrict__ w1,
                        const float* __restrict__ b1, const float* __restrict__ w2,
                        const float* __restrict__ b2, float* __restrict__ att, int total) {
  int idx = blockIdx.x * blockDim.x + threadIdx.x;
  if (idx >= total) return;
  int b = idx / kL, l = idx - b * kL;
  const float* sp = stat + (size_t)b * 4 * kL + l;
  float s0 = sp[0], s1 = sp[kL], s2 = sp[2 * kL], s3 = sp[3 * kL];
  float a0 = b2[0], a1 = b2[1];
  for (int j = 0; j < 16; ++j) {
    float h = fmaxf(w1[j * 4 + 0] * s0 + w1[j * 4 + 1] * s1 + w1[j * 4 + 2] * s2 +
                        w1[j * 4 + 3] * s3 + b1[j],
                    0.f);
    a0 += w2[j] * h;
    a1 += w2[16 + j] * h;
  }
  att[(size_t)b * 2 * kL + l] = sigm(a0);
  att[(size_t)b * 2 * kL + kL + l] = sigm(a1);
}

// Build the 4 gated tensors.
__global__ void gate_inputs(const float* __restrict__ img_n, const float* __restrict__ ev_n,
                            const float* __restrict__ att_cca, const float* __restrict__ csaa,
                            float* __restrict__ img_cca, float* __restrict__ ev_cca,
                            float* __restrict__ img_csa, float* __restrict__ ev_csa, int total) {
  int idx = blockIdx.x * blockDim.x + threadIdx.x;
  if (idx >= total) return;
  int b = idx / (kC * kL);
  int r = idx - b * kC * kL;
  int c = r / kL, p = r - c * kL;
  float xi = img_n[idx], xe = ev_n[idx];
  int odd = c & 1;
  float si = odd ? xe : xi, se = odd ? xi : xe;
  img_cca[idx] = si * att_cca[(0 * kB + b) * kC + c];
  ev_cca[idx] = se * att_cca[(1 * kB + b) * kC + c];
  img_csa[idx] = xi * csaa[(size_t)b * 2 * kL + kL + p];  // att[:,1]
  ev_csa[idx] = xe * csaa[(size_t)b * 2 * kL + p];        // att[:,0]
}

// Causal depthwise conv1d (k=3) + silu.
__global__ void dwconv1d_silu(const float* __restrict__ X, const float* __restrict__ w,
                              const float* __restrict__ bias, float* __restrict__ Y, int total) {
  int idx = blockIdx.x * blockDim.x + threadIdx.x;
  if (idx >= total) return;
  int l = idx % kL;
  int d = (idx / kL) % kD;
  float acc = bias[d];
  const float* wp = w + d * 3;
  for (int t = 0; t < 3; ++t) {
    int ls = l - 2 + t;
    if (ls >= 0) acc += wp[t] * X[idx - l + ls];
  }
  Y[idx] = silu(acc);
}

// Depthwise conv2d 3x3 SAME + silu.
__global__ void dwconv2d_silu(const float* __restrict__ X, const float* __restrict__ w,
                              const float* __restrict__ bias, float* __restrict__ Y, int total) {
  int idx = blockIdx.x * blockDim.x + threadIdx.x;
  if (idx >= total) return;
  int l = idx % kL;
  int d = (idx / kL) % kD;
  int h = l / kW, x = l - h * kW;
  float acc = bias[d];
  const float* wp = w + d * 9;
  for (int ky = 0; ky < 3; ++ky) {
    int hh = h + ky - 1;
    if (hh < 0 || hh >= kH) continue;
    for (int kx = 0; kx < 3; ++kx) {
      int xx = x + kx - 1;
      if (xx < 0 || xx >= kW) continue;
      acc += wp[ky * 3 + kx] * X[idx - l + hh * kW + xx];
    }
  }
  Y[idx] = silu(acc);
}

// Expand (B,D,L) into 4 scan directions (B,4,D,L).
__global__ void expand4(const float* __restrict__ X, float* __restrict__ xs, int total) {
  int idx = blockIdx.x * blockDim.x + threadIdx.x;
  if (idx >= total) return;
  int l = idx % kL;
  int d = (idx / kL) % kD;
  int b = idx / (kD * kL);
  int h = l / kW, x = l - h * kW;
  int lt = x * kH + h;
  float v = X[idx];
  size_t base = (((size_t)b * 4) * kD + d) * kL;
  size_t ks = (size_t)kD * kL;
  xs[base + l] = v;
  xs[base + ks + lt] = v;
  xs[base + 2 * ks + (kL - 1 - l)] = v;
  xs[base + 3 * ks + (kL - 1 - lt)] = v;
}

// Combine 4 scan direction outputs back to (B,D,L).
__global__ void combine4(const float* __restrict__ ys, float* __restrict__ Y, int total) {
  int idx = blockIdx.x * blockDim.x + threadIdx.x;
  if (idx >= total) return;
  int l = idx % kL;
  int d = (idx / kL) % kD;
  int b = idx / (kD * kL);
  int h = l / kW, x = l - h * kW;
  int lt = x * kH + h;
  size_t base = (((size_t)b * 4) * kD + d) * kL;
  size_t ks = (size_t)kD * kL;
  Y[idx] = ys[base + l] + ys[base + ks + lt] + ys[base + 2 * ks + (kL - 1 - l)] +
           ys[base + 3 * ks + (kL - 1 - lt)];
}

// Selective scan. grid = B*Kp blocks of 1024. tid = slot*16 + n; d = slot + 64*j.
__global__ __launch_bounds__(1024) void sel_scan(
    const float* __restrict__ u, const float* __restrict__ dtraw,
    const float* __restrict__ xdbl, const float* __restrict__ A_log,
    const float* __restrict__ Ds, const float* __restrict__ dt_bias,
    float* __restrict__ y, int Kp) {
  const int z = blockIdx.x;
  const int k = z % Kp;
  const int tid = threadIdx.x;
  const int n = tid & 15;
  const int slot = tid >> 4;  // 0..63
  const float* xb = xdbl + (size_t)z * kXRp * kL;
  float h[3], Aa[3], Dv[3], dtb[3];
  for (int j = 0; j < 3; ++j) {
    int d = slot + 64 * j;
    h[j] = 0.f;
    Aa[j] = -expf(A_log[((size_t)k * kD + d) * kN + n]);
    Dv[j] = Ds[k * kD + d];
    dtb[j] = dt_bias[k * kD + d];
  }
  for (int l = 0; l < kL; ++l) {
    float Bv = xb[(size_t)(6 + n) * kL + l];
    float Cv = xb[(size_t)(22 + n) * kL + l];
    for (int j = 0; j < 3; ++j) {
      int d = slot + 64 * j;
      size_t o = ((size_t)z * kD + d) * kL + l;
      float dt = softplusf(dtraw[o] + dtb[j]);
      float uv = u[o];
      h[j] = expf(dt * Aa[j]) * h[j] + dt * Bv * uv;
      float part = Cv * h[j];
      part += __shfl_xor(part, 1, 16);
      part += __shfl_xor(part, 2, 16);
      part += __shfl_xor(part, 4, 16);
      part += __shfl_xor(part, 8, 16);
      if (n == 0) y[o] = part + Dv[j] * uv;
    }
  }
}

// Channel attention MLP (96 -> 6 -> 96), shared for avg+max.
__global__ void ca_mlp(const float* __restrict__ camean, const float* __restrict__ camax,
                       const float* __restrict__ w1, const float* __restrict__ w2,
                       float* __restrict__ att) {
  int b = blockIdx.x;
  int c = threadIdx.x;
  __shared__ float av[kC];
  __shared__ float mxv[kC];
  __shared__ float hs[6];
  av[c] = camean[b * kC + c];
  mxv[c] = camax[b * kC + c];
  __syncthreads();
  if (c < 6) {
    float sa = 0.f, sm = 0.f;
    for (int i = 0; i < kC; ++i) {
      sa += w1[c * kC + i] * av[i];
      sm += w1[c * kC + i] * mxv[i];
    }
    hs[c] = fmaxf(sa, 0.f) + fmaxf(sm, 0.f);
  }
  __syncthreads();
  float s = 0.f;
  for (int j = 0; j < 6; ++j) s += w2[c * 6 + j] * hs[j];
  att[b * kC + c] = sigm(s);
}

// 7x7 spatial attention conv on (B,2,L) stats.
__global__ void sa_conv(const float* __restrict__ stat, const float* __restrict__ w,
                        float* __restrict__ att, int total) {
  int idx = blockIdx.x * blockDim.x + threadIdx.x;
  if (idx >= total) return;
  int b = idx / kL, l = idx - b * kL;
  int h = l / kW, x = l - h * kW;
  float acc = 0.f;
  for (int c2 = 0; c2 < 2; ++c2) {
    const float* sp = stat + (size_t)b * 2 * kL + (size_t)c2 * kL;
    for (int ky = 0; ky < 7; ++ky) {
      int hh = h + ky - 3;
      if (hh < 0 || hh >= kH) continue;
      for (int kx = 0; kx < 7; ++kx) {
        int xx = x + kx - 3;
        if (xx < 0 || xx >= kW) continue;
        acc += w[(c2 * 7 + ky) * 7 + kx] * sp[hh * kW + xx];
      }
    }
  }
  att[(size_t)b * kL + l] = sigm(acc);
}

__global__ void scale_chan(float* __restrict__ cat, const float* __restrict__ att, int rowoff,
                           int total) {
  int idx = blockIdx.x * blockDim.x + threadIdx.x;
  if (idx >= total) return;
  int b = idx / (kC * kL);
  int r = idx - b * kC * kL;
  int c = r / kL, p = r - c * kL;
  cat[(size_t)b * 2 * kC * kL + (size_t)(rowoff + c) * kL + p] *= att[b * kC + c];
}

__global__ void scale_pix(float* __restrict__ cat, const float* __restrict__ att, int rowoff,
                          int total) {
  int idx = blockIdx.x * blockDim.x + threadIdx.x;
  if (idx >= total) return;
  int b = idx / (kC * kL);
  int r = idx - b * kC * kL;
  int c = r / kL, p = r - c * kL;
  cat[(size_t)b * 2 * kC * kL + (size_t)(rowoff + c) * kL + p] *= att[(size_t)b * kL + p];
}

__global__ void copyk(const float* __restrict__ x, float* __restrict__ y, int n) {
  int i = blockIdx.x * blockDim.x + threadIdx.x;
  if (i < n) y[i] = x[i];
}

__global__ void pad_xproj(const float* __restrict__ src, float* __restrict__ dst, int Kp) {
  int idx = blockIdx.x * blockDim.x + threadIdx.x;
  int total = Kp * kXRp * kD;
  if (idx >= total) return;
  int k = idx / (kXRp * kD);
  int r = (idx / kD) % kXRp;
  int d = idx % kD;
  dst[idx] = (r < kXR) ? src[((size_t)k * kXR + r) * kD + d] : 0.f;
}

__global__ void pad_dtproj(const float* __restrict__ src, float* __restrict__ dst, int Kp) {
  int idx = blockIdx.x * blockDim.x + threadIdx.x;
  int total = Kp * kD * kRp;
  if (idx >= total) return;
  int k = idx / (kD * kRp);
  int d = (idx / kRp) % kD;
  int r = idx % kRp;
  dst[idx] = (r < 6) ? src[((size_t)k * kD + d) * 6 + r] : 0.f;
}

// ---------------------------------------------------------------------------
// Host side
// ---------------------------------------------------------------------------
enum {
  IN_XIMG = 0, IN_XEV = 1,
  P_NORM_I_W = 2, P_NORM_I_B = 3, P_NORM_E_W = 4, P_NORM_E_B = 5,
  P_CCA_W1 = 6, P_CCA_W2 = 7,
  P_CSA_W1 = 8, P_CSA_B1 = 9, P_CSA_W2 = 10, P_CSA_B2 = 11,
  P_CA1_W1 = 12, P_CA1_W2 = 13, P_CA2_W1 = 14, P_CA2_W2 = 15,
  P_SA1_W = 16, P_SA2_W = 17,
  P_SSM_I = 18, P_SSM_E = 31, P_SS2D_I = 44, P_SS2D_E = 57,
  P_CI1_W = 70, P_CI1_B = 71, P_CE1_W = 72, P_CE1_B = 73,
  P_CI2_W = 74, P_CI2_B = 75, P_CE2_W = 76, P_CE2_B = 77,
  P_CI3_W = 78, P_CI3_B = 79, P_CE3_W = 80, P_CE3_B = 81,
  N_PARAMS = 82
};
enum {
  S_INW = 0, S_INB = 1, S_CONVW = 2, S_CONVB = 3, S_XPROJ = 4, S_DTPROJ = 5,
  S_DTBIAS = 6, S_ALOG = 7, S_DS = 8, S_NORMW = 9, S_NORMB = 10, S_OUTW = 11, S_OUTB = 12
};

struct Bufs {
  float *bufA, *bufU, *bufXS, *bufXD, *bufDT, *bufYS, *bufYC;
  float *pwx, *pwdt;
  float *camean, *camax, *caatt;
  float *sastat, *saatt;
};

static void run_temporal(const float* const* S, const float* G, const float* ciw,
                         const float* cib, const float* caw1, const float* caw2, float* cat,
                         const Bufs& bf, hipStream_t stream) {
  const long CL = (long)kC * kL, DL = (long)kD * kL, C2L = 2L * kC * kL;
  // in_proj (D x C)
  gemm_wmma_f32<<<dim3(kL / 64, kD / 16, kB), 32, 0, stream>>>(
      S[S_INW], G, S[S_INB], bf.bufA, kD, kC, kL, 1, 0, CL, DL, 0);
  int tot = kB * kD * kL;
  dwconv1d_silu<<<CDIV(tot, 256), 256, 0, stream>>>(bf.bufA, S[S_CONVW], S[S_CONVB], bf.bufU, tot);
  pad_xproj<<<CDIV(kXRp * kD, 256), 256, 0, stream>>>(S[S_XPROJ], bf.pwx, 1);
  gemm_wmma_f32<<<dim3(kL / 64, kXRp / 16, kB), 32, 0, stream>>>(
      bf.pwx, bf.bufU, nullptr, bf.bufXD, kXRp, kD, kL, 1, (long)kXRp * kD, DL, (long)kXRp * kL, 0);
  pad_dtproj<<<CDIV(kD * kRp, 256), 256, 0, stream>>>(S[S_DTPROJ], bf.pwdt, 1);
  gemm_wmma_f32<<<dim3(kL / 64, kD / 16, kB), 32, 0, stream>>>(
      bf.pwdt, bf.bufXD, nullptr, bf.bufDT, kD, kRp, kL, 1, (long)kD * kRp, (long)kXRp * kL, DL, 0);
  sel_scan<<<kB, 1024, 0, stream>>>(bf.bufU, bf.bufDT, bf.bufXD, S[S_ALOG], S[S_DS],
                                    S[S_DTBIAS], bf.bufYS, 1);
  layernorm_rows<<<CDIV(kB * kL, 256), 256, 0, stream>>>(bf.bufYS, bf.bufYC, S[S_NORMW],
                                                         S[S_NORMB], kD, kL, kB);
  // shortcut conv1x1 into cat rows 0..95, then out_proj accumulates
  gemm_wmma_f32<<<dim3(kL / 64, kC / 16, kB), 32, 0, stream>>>(
      ciw, G, cib, cat, kC, kC, kL, 1, 0, CL, C2L, 0);
  gemm_wmma_f32<<<dim3(kL / 64, kC / 16, kB), 32, 0, stream>>>(
      S[S_OUTW], bf.bufYC, S[S_OUTB], cat, kC, kD, kL, 1, 0, DL, C2L, 1);
  // channel attention on xic (rows 0..95)
  rowstat<<<kB * kC, 256, 0, stream>>>(cat, bf.camean, bf.camax, kL, kC, C2L, (long)kL);
  ca_mlp<<<kB, kC, 0, stream>>>(bf.camean, bf.camax, caw1, caw2, bf.caatt);
  scale_chan<<<CDIV(kB * kC * kL, 256), 256, 0, stream>>>(cat, bf.caatt, 0, kB * kC * kL);
}

static void run_ss2d(const float* const* S, const float* G, const float* ciw, const float* cib,
                     const float* saw, float* cat, const Bufs& bf, hipStream_t stream) {
  const long CL = (long)kC * kL, DL = (long)kD * kL, C2L = 2L * kC * kL;
  const int Kp = 4;
  gemm_wmma_f32<<<dim3(kL / 64, kD / 16, kB), 32, 0, stream>>>(
      S[S_INW], G, S[S_INB], bf.bufA, kD, kC, kL, 1, 0, CL, DL, 0);
  int tot = kB * kD * kL;
  dwconv2d_silu<<<CDIV(tot, 256), 256, 0, stream>>>(bf.bufA, S[S_CONVW], S[S_CONVB], bf.bufU, tot);
  expand4<<<CDIV(tot, 256), 256, 0, stream>>>(bf.bufU, bf.bufXS, tot);
  pad_xproj<<<CDIV(Kp * kXRp * kD, 256), 256, 0, stream>>>(S[S_XPROJ], bf.pwx, Kp);
  gemm_wmma_f32<<<dim3(kL / 64, kXRp / 16, kB * Kp), 32, 0, stream>>>(
      bf.pwx, bf.bufXS, nullptr, bf.bufXD, kXRp, kD, kL, Kp, (long)kXRp * kD, DL,
      (long)kXRp * kL, 0);
  pad_dtproj<<<CDIV(Kp * kD * kRp, 256), 256, 0, stream>>>(S[S_DTPROJ], bf.pwdt, Kp);
  gemm_wmma_f32<<<dim3(kL / 64, kD / 16, kB * Kp), 32, 0, stream>>>(
      bf.pwdt, bf.bufXD, nullptr, bf.bufDT, kD, kRp, kL, Kp, (long)kD * kRp, (long)kXRp * kL,
      DL, 0);
  sel_scan<<<kB * Kp, 1024, 0, stream>>>(bf.bufXS, bf.bufDT, bf.bufXD, S[S_ALOG], S[S_DS],
                                         S[S_DTBIAS], bf.bufYS, Kp);
  combine4<<<CDIV(tot, 256), 256, 0, stream>>>(bf.bufYS, bf.bufYC, tot);
  layernorm_rows<<<CDIV(kB * kL, 256), 256, 0, stream>>>(bf.bufYC, bf.bufA, S[S_NORMW],
                                                         S[S_NORMB], kD, kL, kB);
  // shortcut conv1x1 into cat rows 96..191, then out_proj accumulates
  gemm_wmma_f32<<<dim3(kL / 64, kC / 16, kB), 32, 0, stream>>>(
      ciw, G, cib, cat + (size_t)kC * kL, kC, kC, kL, 1, 0, CL, C2L, 0);
  gemm_wmma_f32<<<dim3(kL / 64, kC / 16, kB), 32, 0, stream>>>(
      S[S_OUTW], bf.bufA, S[S_OUTB], cat + (size_t)kC * kL, kC, kD, kL, 1, 0, DL, C2L, 1);
  // spatial attention on xis (rows 96..191)
  colstat<<<CDIV(kB * kL, 256), 256, 0, stream>>>(cat + (size_t)kC * kL, bf.sastat,
                                                  bf.sastat + kL, kC, C2L, 2L * kL, kB * kL);
  sa_conv<<<CDIV(kB * kL, 256), 256, 0, stream>>>(bf.sastat, saw, bf.saatt, kB * kL);
  scale_pix<<<CDIV(kB * kC * kL, 256), 256, 0, stream>>>(cat, bf.saatt, kC, kB * kC * kL);
}

extern "C" void kernel_launch(void* const* d_in, const int* in_sizes, int n_in, void* d_out,
                              int out_size, void* d_ws, size_t ws_size, hipStream_t stream) {
  (void)in_sizes;
  (void)out_size;
  (void)ws_size;
  const float* P[N_PARAMS];
  for (int i = 0; i < N_PARAMS && i < n_in; ++i) P[i] = (const float*)d_in[i];
  float* out = (float*)d_out;
  float* ws = (float*)d_ws;

  const size_t BCL = (size_t)kB * kC * kL;
  const size_t BDL = (size_t)kB * kD * kL;
  size_t off = 0;
  auto alloc = [&](size_t n) {
    float* p = ws + off;
    off += n;
    return p;
  };
  float* img_n = alloc(BCL);
  float* ev_n = alloc(BCL);
  float* img_cca = alloc(BCL);
  float* ev_cca = alloc(BCL);
  float* img_csa = alloc(BCL);
  float* ev_csa = alloc(BCL);
  float* cat_i = alloc(2 * BCL);
  float* cat_e = alloc(2 * BCL);
  float* gmean_i = alloc(kB * kC);
  float* gmean_e = alloc(kB * kC);
  float* att_cca = alloc(2 * kB * kC);
  float* csastat = alloc((size_t)kB * 4 * kL);
  float* csaatt = alloc((size_t)kB * 2 * kL);
  Bufs bf;
  bf.camean = alloc(kB * kC);
  bf.camax = alloc(kB * kC);
  bf.caatt = alloc(kB * kC);
  bf.sastat = alloc((size_t)kB * 2 * kL);
  bf.saatt = alloc((size_t)kB * kL);
  bf.pwx = alloc((size_t)4 * kXRp * kD);
  bf.pwdt = alloc((size_t)4 * kD * kRp);
  bf.bufA = alloc(BDL);
  bf.bufU = alloc(BDL);
  bf.bufXS = alloc(4 * BDL);
  bf.bufXD = alloc((size_t)kB * 4 * kXRp * kL);
  bf.bufDT = alloc(4 * BDL);
  bf.bufYS = alloc(4 * BDL);
  bf.bufYC = alloc(BDL);

  // ---- Stage 1: channel layernorms ----
  layernorm_rows<<<CDIV(kB * kL, 256), 256, 0, stream>>>(P[IN_XIMG], img_n, P[P_NORM_I_W],
                                                         P[P_NORM_I_B], kC, kL, kB);
  layernorm_rows<<<CDIV(kB * kL, 256), 256, 0, stream>>>(P[IN_XEV], ev_n, P[P_NORM_E_W],
                                                         P[P_NORM_E_B], kC, kL, kB);
  // ---- CCA: channel means + gate MLPs ----
  rowstat<<<kB * kC, 256, 0, stream>>>(img_n, gmean_i, nullptr, kL, kC, (long)kC * kL, (long)kL);
  rowstat<<<kB * kC, 256, 0, stream>>>(ev_n, gmean_e, nullptr, kL, kC, (long)kC * kL, (long)kL);
  cca_gate<<<2 * kB, kC, 0, stream>>>(gmean_i, gmean_e, P[P_CCA_W1], P[P_CCA_W2], att_cca);
  // ---- CSA: per-pixel stats + MLP ----
  colstat<<<CDIV(kB * kL, 256), 256, 0, stream>>>(img_n, csastat, csastat + kL, kC,
                                                  (long)kC * kL, 4L * kL, kB * kL);
  colstat<<<CDIV(kB * kL, 256), 256, 0, stream>>>(ev_n, csastat + 2 * kL, csastat + 3 * kL, kC,
                                                  (long)kC * kL, 4L * kL, kB * kL);
  csa_att<<<CDIV(kB * kL, 256), 256, 0, stream>>>(csastat, P[P_CSA_W1], P[P_CSA_B1],
                                                  P[P_CSA_W2], P[P_CSA_B2], csaatt, kB * kL);
  // ---- Gated inputs ----
  gate_inputs<<<CDIV((int)BCL, 256), 256, 0, stream>>>(img_n, ev_n, att_cca, csaatt, img_cca,
                                                       ev_cca, img_csa, ev_csa, (int)BCL);

  // ---- IMG stream ----
  run_temporal(&P[P_SSM_I], img_cca, P[P_CI1_W], P[P_CI1_B], P[P_CA1_W1], P[P_CA1_W2], cat_i,
               bf, stream);
  run_ss2d(&P[P_SS2D_I], img_csa, P[P_CI2_W], P[P_CI2_B], P[P_SA1_W], cat_i, bf, stream);
  // ---- EV stream ----
  run_temporal(&P[P_SSM_E], ev_cca, P[P_CE1_W], P[P_CE1_B], P[P_CA2_W1], P[P_CA2_W2], cat_e,
               bf, stream);
  run_ss2d(&P[P_SS2D_E], ev_csa, P[P_CE2_W], P[P_CE2_B], P[P_SA2_W], cat_e, bf, stream);

  // ---- Final fuse: out = x + ci3 * [xic; xis] ----
  copyk<<<CDIV((int)BCL, 256), 256, 0, stream>>>(P[IN_XIMG], out, (int)BCL);
  gemm_wmma_f32<<<dim3(kL / 64, kC / 16, kB), 32, 0, stream>>>(
      P[P_CI3_W], cat_i, P[P_CI3_B], out, kC, 2 * kC, kL, 1, 0, 2L * kC * kL, (long)kC * kL, 1);
  copyk<<<CDIV((int)BCL, 256), 256, 0, stream>>>(P[IN_XEV], out + BCL, (int)BCL);
  gemm_wmma_f32<<<dim3(kL / 64, kC / 16, kB), 32, 0, stream>>>(
      P[P_CE3_W], cat_e, P[P_CE3_B], out + BCL, kC, 2 * kC, kL, 1, 0, 2L * kC * kL,
      (long)kC * kL, 1);
}